// SimpleGraphSageEdgeLayer_83476984365556
// MI455X (gfx1250) — compile-verified
//
#include <hip/hip_runtime.h>

typedef __attribute__((ext_vector_type(2))) float v2f;
typedef __attribute__((ext_vector_type(8))) float v8f;

#define IN_DIM 64
#define OUT_DIM 64
#define WPAD (IN_DIM + 2)        // padded LDS row: 264 B stride, 8 B aligned, bank-spread
#define ENC_NEG_INF 0x007FFFFFu  // order-preserving encoding of -inf

// Order-preserving float->uint mapping: a <= b  <=>  enc(a) <= enc(b)
__device__ __forceinline__ unsigned encOrd(float f) {
  unsigned u = __float_as_uint(f);
  return (u & 0x80000000u) ? ~u : (u | 0x80000000u);
}
__device__ __forceinline__ float decOrd(unsigned e) {
  return (e & 0x80000000u) ? __uint_as_float(e & 0x7FFFFFFFu)
                           : __uint_as_float(~e);
}

// ---------------------------------------------------------------------------
// Kernel 1: initialize encoded mailbox to enc(-inf)
// ---------------------------------------------------------------------------
__global__ __launch_bounds__(256) void init_mailbox_kernel(unsigned* __restrict__ cEnc, int n) {
  int i = blockIdx.x * 256 + threadIdx.x;
  if (i < n) cEnc[i] = ENC_NEG_INF;
}

// ---------------------------------------------------------------------------
// Kernel 2: hn = h*norm; Ah = hn@A_w + A_b; Bh = hn@B_w + B_b
// One wave32 per 16-node tile; WMMA F32_16x16x4_F32, K=64 in 16 steps,
// 4 N-tiles of 16 columns, for both weight matrices (128 WMMAs / wave).
// Weights staged TRANSPOSED in LDS so each B fragment is one aligned b64 load.
// ---------------------------------------------------------------------------
__global__ __launch_bounds__(128)
void node_gemm_kernel(const float* __restrict__ h, const float* __restrict__ norm,
                      const float* __restrict__ Aw, const float* __restrict__ Ab,
                      const float* __restrict__ Bw, const float* __restrict__ Bb,
                      float* __restrict__ hn, float* __restrict__ Ah, float* __restrict__ Bh,
                      int nNodes, int nTiles) {
  __shared__ float sWt[2][OUT_DIM][WPAD];  // ~33 KB: A_w^T, B_w^T (padded rows)
  __shared__ float sH[4][16][IN_DIM];      // 16 KB: per-wave hn tiles

  const int tid  = threadIdx.x;
  const int wave = tid >> 5;
  const int lane = tid & 31;

  // ---- stage both weight matrices transposed into LDS ----
  // Global read: coalesced float4 along the OUT dimension; LDS write: sWt[n][k].
  {
    const float4* a4 = (const float4*)Aw;
    const float4* b4 = (const float4*)Bw;
    for (int i = tid; i < IN_DIM * OUT_DIM / 4; i += 128) {
      const int k = i >> 4;            // row of W (= K index)
      const int n = (i & 15) * 4;      // starting column
      float4 a = a4[i];
      float4 b = b4[i];
      sWt[0][n + 0][k] = a.x; sWt[0][n + 1][k] = a.y;
      sWt[0][n + 2][k] = a.z; sWt[0][n + 3][k] = a.w;
      sWt[1][n + 0][k] = b.x; sWt[1][n + 1][k] = b.y;
      sWt[1][n + 2][k] = b.z; sWt[1][n + 3][k] = b.w;
    }
  }

  // ---- stage this wave's 16x64 hn tile (h * norm), also spill hn to global ----
  const int tile  = blockIdx.x * 4 + wave;
  const int tileC = (tile < nTiles) ? tile : (nTiles - 1);
  const int base  = tileC * 16;
#pragma unroll
  for (int i = 0; i < 8; ++i) {
    int flat = i * 32 + lane;        // float4 units; 256 per tile
    int row  = flat >> 4;
    int c4   = flat & 15;
    int node  = base + row;
    int nodeC = (node < nNodes) ? node : (nNodes - 1);
    float4 v = *(const float4*)(h + (size_t)nodeC * IN_DIM + c4 * 4);
    float nm = norm[nodeC];
    v.x *= nm; v.y *= nm; v.z *= nm; v.w *= nm;
    *(float4*)&sH[wave][row][c4 * 4] = v;
    if (tile < nTiles && node < nNodes)
      *(float4*)(hn + (size_t)node * IN_DIM + c4 * 4) = v;
  }
  __syncthreads();  // all 128 threads reach this barrier

  // ---- WMMA: every lane active (EXEC all ones required by WMMA) ----
  const int half = lane >> 4;   // which half-wave
  const int mn   = lane & 15;   // M index for A frag / N index for B frag
  v8f accA[4] = {{}, {}, {}, {}};
  v8f accB[4] = {{}, {}, {}, {}};

#pragma unroll 4
  for (int kk = 0; kk < 16; ++kk) {
    const int k0 = kk * 4 + half * 2;  // even -> 8B aligned LDS reads
    v2f a = *(const v2f*)&sH[wave][mn][k0];
#pragma unroll
    for (int nt = 0; nt < 4; ++nt) {
      const int col = nt * 16 + mn;
      // Transposed layout: fragment = contiguous pair -> single ds_load_b64
      v2f bA = *(const v2f*)&sWt[0][col][k0];
      v2f bB = *(const v2f*)&sWt[1][col][k0];
      accA[nt] = __builtin_amdgcn_wmma_f32_16x16x4_f32(
          false, a, false, bA, (short)0, accA[nt], false, false);
      accB[nt] = __builtin_amdgcn_wmma_f32_16x16x4_f32(
          false, a, false, bB, (short)0, accB[nt], false, false);
    }
  }

  if (tile >= nTiles) return;  // wave-uniform; no barriers after this point

  // ---- add bias, store. C/D layout: VGPR r -> M = r + 8*half, N = lane&15 ----
  if (base + 16 <= nNodes) {
    // Fast path (every tile except possibly the last): no per-element
    // predication -> straight-line global_store_b32 runs, no exec-mask churn.
#pragma unroll
    for (int nt = 0; nt < 4; ++nt) {
      const int col = nt * 16 + mn;
      const float ab = Ab[col];
      const float bb = Bb[col];
      float* pA = Ah + (size_t)(base + half * 8) * OUT_DIM + col;
      float* pB = Bh + (size_t)(base + half * 8) * OUT_DIM + col;
#pragma unroll
      for (int r = 0; r < 8; ++r) {
        pA[(size_t)r * OUT_DIM] = accA[nt][r] + ab;
        pB[(size_t)r * OUT_DIM] = accB[nt][r] + bb;
      }
    }
  } else {
    // Tail tile: per-row predication.
#pragma unroll
    for (int nt = 0; nt < 4; ++nt) {
      const int col = nt * 16 + mn;
      const float ab = Ab[col];
      const float bb = Bb[col];
#pragma unroll
      for (int r = 0; r < 8; ++r) {
        const int node = base + r + half * 8;
        if (node < nNodes) {
          Ah[(size_t)node * OUT_DIM + col] = accA[nt][r] + ab;
          Bh[(size_t)node * OUT_DIM + col] = accB[nt][r] + bb;
        }
      }
    }
  }
}

// ---------------------------------------------------------------------------
// Kernel 3: per-edge gating + segment-max via ordered-uint atomicMax.
// 16 threads per edge, float4 gathers (Ah/Bh are L2-resident: 25.6 MB total).
// ---------------------------------------------------------------------------
__global__ __launch_bounds__(256)
void edge_kernel(const int* __restrict__ src, const int* __restrict__ dst,
                 const float* __restrict__ Ah, const float* __restrict__ Bh,
                 unsigned* __restrict__ cEnc, int nEdges) {
  const int t = blockIdx.x * 256 + threadIdx.x;
  const int e = t >> 4;
  if (e >= nEdges) return;

  // Streaming prefetch of upcoming edge indices (global_prefetch_b8);
  // one lane per block, well ahead of the current window.
  if (threadIdx.x == 0) {
    __builtin_prefetch(src + e + 4096, 0, 1);
    __builtin_prefetch(dst + e + 4096, 0, 1);
  }

  const int q = (t & 15) * 4;
  const int s = src[e];
  const int d = dst[e];
  const float4 bs = *(const float4*)(Bh + (size_t)s * OUT_DIM + q);
  const float4 bd = *(const float4*)(Bh + (size_t)d * OUT_DIM + q);
  const float4 as = *(const float4*)(Ah + (size_t)s * OUT_DIM + q);

  const float m0 = as.x / (1.0f + __expf(-(bs.x + bd.x)));
  const float m1 = as.y / (1.0f + __expf(-(bs.y + bd.y)));
  const float m2 = as.z / (1.0f + __expf(-(bs.z + bd.z)));
  const float m3 = as.w / (1.0f + __expf(-(bs.w + bd.w)));

  unsigned* cp = cEnc + (size_t)d * OUT_DIM + q;
  atomicMax(cp + 0, encOrd(m0));
  atomicMax(cp + 1, encOrd(m1));
  atomicMax(cp + 2, encOrd(m2));
  atomicMax(cp + 3, encOrd(m3));
}

// ---------------------------------------------------------------------------
// Kernel 4: decode mailbox, zero isolated nodes, concat, L2-normalize, *norm.
// One wave32 per node; wave shuffle reduction (warpSize == 32 on gfx1250).
// ---------------------------------------------------------------------------
__global__ __launch_bounds__(256)
void finalize_kernel(const float* __restrict__ hn, const unsigned* __restrict__ cEnc,
                     const float* __restrict__ norm, float* __restrict__ out, int nNodes) {
  const int node = (blockIdx.x * 256 + threadIdx.x) >> 5;
  const int lane = threadIdx.x & 31;
  if (node >= nNodes) return;

  const float2 h2 = *(const float2*)(hn + (size_t)node * IN_DIM + lane * 2);
  const uint2 ce = *(const uint2*)(cEnc + (size_t)node * OUT_DIM + lane * 2);
  float c0 = decOrd(ce.x);
  float c1 = decOrd(ce.y);
  if (!(fabsf(c0) < __builtin_inff())) c0 = 0.0f;  // -inf (no edges) or NaN -> 0
  if (!(fabsf(c1) < __builtin_inff())) c1 = 0.0f;

  float s = h2.x * h2.x + h2.y * h2.y + c0 * c0 + c1 * c1;
#pragma unroll
  for (int off = 16; off > 0; off >>= 1) s += __shfl_xor(s, off, 32);

  const float scale = norm[node] / fmaxf(sqrtf(s), 1e-12f);
  float* o = out + (size_t)node * (IN_DIM + OUT_DIM);
  o[lane * 2 + 0]          = h2.x * scale;
  o[lane * 2 + 1]          = h2.y * scale;
  o[IN_DIM + lane * 2 + 0] = c0 * scale;
  o[IN_DIM + lane * 2 + 1] = c1 * scale;
}

// ---------------------------------------------------------------------------
extern "C" void kernel_launch(void* const* d_in, const int* in_sizes, int n_in,
                              void* d_out, int out_size, void* d_ws, size_t ws_size,
                              hipStream_t stream) {
  const float* h    = (const float*)d_in[0];
  const float* norm = (const float*)d_in[1];
  const int*   src  = (const int*)d_in[2];
  const int*   dst  = (const int*)d_in[3];
  const float* Aw   = (const float*)d_in[4];
  const float* Ab   = (const float*)d_in[5];
  const float* Bw   = (const float*)d_in[6];
  const float* Bb   = (const float*)d_in[7];
  float* out = (float*)d_out;

  const int nNodes = in_sizes[1];   // norm is [N,1]
  const int nEdges = in_sizes[2];

  // workspace partition: hn | Ah | Bh | cEnc  (each N*64 words)
  const size_t nodeF = (size_t)nNodes * OUT_DIM;
  float*    hn   = (float*)d_ws;
  float*    Ah   = hn + nodeF;
  float*    Bh   = Ah + nodeF;
  unsigned* cEnc = (unsigned*)(Bh + nodeF);

  const int nTiles = (nNodes + 15) / 16;

  // 1) mailbox init
  {
    int n = (int)nodeF;
    init_mailbox_kernel<<<(n + 255) / 256, 256, 0, stream>>>(cEnc, n);
  }
  // 2) node GEMMs via WMMA (4 waves = 4 node-tiles per block)
  node_gemm_kernel<<<(nTiles + 3) / 4, 128, 0, stream>>>(
      h, norm, Aw, Ab, Bw, Bb, hn, Ah, Bh, nNodes, nTiles);
  // 3) edge gating + atomic segment-max (16 threads/edge)
  {
    long long threads = (long long)nEdges * 16;
    int blocks = (int)((threads + 255) / 256);
    edge_kernel<<<blocks, 256, 0, stream>>>(src, dst, Ah, Bh, cEnc, nEdges);
  }
  // 4) finalize (8 nodes per 256-thread block)
  finalize_kernel<<<(nNodes + 7) / 8, 256, 0, stream>>>(hn, cEnc, norm, out, nNodes);
}